// VectorQuantizerEMA_17008070492204
// MI455X (gfx1250) — compile-verified
//
#include <hip/hip_runtime.h>

#define K_CODES 1024
#define D_DIM   256
#define N_ROWS  (16 * 4096)
#define BETA_C  0.25f
#define DECAY_C 0.99f
#define EPS_C   1e-05f

typedef __attribute__((ext_vector_type(2))) float v2f;
typedef __attribute__((ext_vector_type(8))) float v8f;

// ---------------------------------------------------------------------------
// Kernel 1: per-code squared L2 norm  ||e_k||^2
// ---------------------------------------------------------------------------
__global__ __launch_bounds__(256) void vq_enorm_kernel(
    const float* __restrict__ embed, float* __restrict__ enorm) {
  int k = blockIdx.x * 256 + threadIdx.x;
  if (k >= K_CODES) return;
  const float4* p = (const float4*)(embed + (size_t)k * D_DIM);
  float s = 0.f;
#pragma unroll 8
  for (int i = 0; i < D_DIM / 4; ++i) {
    float4 v = p[i];
    s += v.x * v.x + v.y * v.y + v.z * v.z + v.w * v.w;
  }
  enorm[k] = s;
}

// ---------------------------------------------------------------------------
// Kernel 2: fused distance GEMM (fp32 WMMA) + argmin over K
//   dist(m,k) = ||e_k||^2 - 2 * <z_m, e_k>   (||z_m||^2 constant per row)
//   Each wave owns a 16-row M-tile; A fragments register-resident across the
//   whole K loop; B tiles (16 codes x 256 dims) staged in LDS per block.
// ---------------------------------------------------------------------------
__global__ __launch_bounds__(256) void vq_argmin_kernel(
    const float* __restrict__ z, const float* __restrict__ embed,
    const float* __restrict__ enorm, int* __restrict__ idx_ws,
    float* __restrict__ idx_out) {
  __shared__ float Bt[16 * D_DIM];  // 16 KB: one 16-code tile of the codebook

  const int lane = threadIdx.x & 31;
  const int wave = threadIdx.x >> 5;
  const int l16  = lane & 15;
  const int hi   = lane >> 4;  // 0 or 1: which K-pair of the 16x16x4 frag
  const int m0   = blockIdx.x * 128 + wave * 16;

  // A-matrix fragments (16x4 fp32 per WMMA): lane l<16 holds A[l][K0,K1],
  // lane l+16 holds A[l][K2,K3]  ->  float2 at z[(m0+l16)*D + dd + hi*2]
  v2f afrag[D_DIM / 4];
  const float* pz = z + (size_t)(m0 + l16) * D_DIM + hi * 2;
#pragma unroll
  for (int i = 0; i < D_DIM / 4; ++i)
    afrag[i] = *(const v2f*)(pz + i * 4);

  float best[8];
  int   bidx[8];
#pragma unroll
  for (int r = 0; r < 8; ++r) { best[r] = 3.4e38f; bidx[r] = 0; }

  // B-matrix fragment source inside LDS tile (same striping as A)
  const float* pb = Bt + l16 * D_DIM + hi * 2;

  for (int kt = 0; kt < K_CODES / 16; ++kt) {
    const int k0 = kt * 16;
    __syncthreads();
    {  // cooperative stage of 16 codebook rows into LDS (float4 coalesced)
      const float4* src = (const float4*)(embed + (size_t)k0 * D_DIM);
      float4* dst = (float4*)Bt;
      for (int j = threadIdx.x; j < 16 * D_DIM / 4; j += 256) dst[j] = src[j];
    }
    __syncthreads();

    v8f acc = {0.f, 0.f, 0.f, 0.f, 0.f, 0.f, 0.f, 0.f};
#pragma unroll
    for (int i = 0; i < D_DIM / 4; ++i) {
      v2f b = *(const v2f*)(pb + i * 4);
      acc = __builtin_amdgcn_wmma_f32_16x16x4_f32(
          /*neg_a=*/false, afrag[i], /*neg_b=*/false, b,
          /*c_mod=*/(short)0, acc, /*reuse_a=*/false, /*reuse_b=*/false);
    }

    // C/D layout: VGPR r, lanes 0-15 -> row r, col=lane; lanes 16-31 -> row r+8
    const float en   = enorm[k0 + l16];
    const int   kidx = k0 + l16;
#pragma unroll
    for (int r = 0; r < 8; ++r) {
      float dv = en - 2.f * acc[r];
      if (dv < best[r]) { best[r] = dv; bidx[r] = kidx; }
    }
  }

  // min-reduce across the 16 columns (shfl_xor 8..1 stays inside each half),
  // lowest-index tie-break to match jnp.argmin
#pragma unroll
  for (int r = 0; r < 8; ++r) {
#pragma unroll
    for (int off = 8; off >= 1; off >>= 1) {
      float ov = __shfl_xor(best[r], off, 32);
      int   oi = __shfl_xor(bidx[r], off, 32);
      if (ov < best[r] || (ov == best[r] && oi < bidx[r])) {
        best[r] = ov; bidx[r] = oi;
      }
    }
    if (l16 == 0) {
      int row = m0 + r + hi * 8;
      idx_ws[row]  = bidx[r];
      idx_out[row] = (float)bidx[r];
    }
  }
}

// ---------------------------------------------------------------------------
// Kernel 3: gather quantized rows, straight-through output, loss partials,
//           segment-sum scatter (counts, embed_sum) via fp32 HW atomics
// ---------------------------------------------------------------------------
__global__ __launch_bounds__(256) void vq_gather_scatter_kernel(
    const float* __restrict__ z, const float* __restrict__ embed,
    const int* __restrict__ idx, float* __restrict__ q_out,
    float* __restrict__ counts, float* __restrict__ embed_sum,
    float* __restrict__ sqsum) {
  __shared__ float wsum[8];
  const int row = blockIdx.x;
  const int d   = threadIdx.x;
  const int k   = idx[row];

  const float zv = z[(size_t)row * D_DIM + d];
  const float ev = embed[(size_t)k * D_DIM + d];
  // straight-through estimator, kept in the reference's fp order: z + (q - z)
  q_out[(size_t)row * D_DIM + d] = zv + (ev - zv);

  unsafeAtomicAdd(&embed_sum[(size_t)k * D_DIM + d], zv);

  float diff = zv - ev;
  float s = diff * diff;
#pragma unroll
  for (int off = 16; off >= 1; off >>= 1) s += __shfl_down(s, off, 32);
  if ((threadIdx.x & 31) == 0) wsum[threadIdx.x >> 5] = s;
  __syncthreads();
  if (threadIdx.x == 0) {
    float t = 0.f;
#pragma unroll
    for (int w = 0; w < 8; ++w) t += wsum[w];
    unsafeAtomicAdd(sqsum, t);
    unsafeAtomicAdd(&counts[k], 1.0f);
  }
}

// ---------------------------------------------------------------------------
// Kernel 4: EMA cluster-size update + Sum(new_cluster_size) + loss scalar
// ---------------------------------------------------------------------------
__global__ __launch_bounds__(1024) void vq_finalize_cs_kernel(
    const float* __restrict__ cluster_size, const float* __restrict__ counts,
    const float* __restrict__ sqsum, float* __restrict__ ncs_out,
    float* __restrict__ loss_out, float* __restrict__ nsum_ws) {
  __shared__ float red[32];
  const int k = threadIdx.x;
  float ncs = DECAY_C * cluster_size[k] + (1.f - DECAY_C) * counts[k];
  ncs_out[k] = ncs;
  float s = ncs;
#pragma unroll
  for (int off = 16; off >= 1; off >>= 1) s += __shfl_down(s, off, 32);
  if ((k & 31) == 0) red[k >> 5] = s;
  __syncthreads();
  if (k < 32) {
    float t = red[k];
#pragma unroll
    for (int off = 16; off >= 1; off >>= 1) t += __shfl_down(t, off, 32);
    if (k == 0) {
      nsum_ws[0]  = t;
      loss_out[0] = BETA_C * sqsum[0] / (float)((size_t)N_ROWS * D_DIM);
    }
  }
}

// ---------------------------------------------------------------------------
// Kernel 5: EMA embed_avg update + Laplace-smoothed codebook
// ---------------------------------------------------------------------------
__global__ __launch_bounds__(256) void vq_finalize_embed_kernel(
    const float* __restrict__ embed_avg, const float* __restrict__ embed_sum,
    const float* __restrict__ ncs, const float* __restrict__ nsum_ws,
    float* __restrict__ new_embed, float* __restrict__ new_embed_avg) {
  const int k = blockIdx.x;
  const int d = threadIdx.x;
  const size_t i = (size_t)k * D_DIM + d;
  float na = DECAY_C * embed_avg[i] + (1.f - DECAY_C) * embed_sum[i];
  new_embed_avg[i] = na;
  float n = nsum_ws[0];
  float smooth = (ncs[k] + EPS_C) / (n + (float)K_CODES * EPS_C) * n;
  new_embed[i] = na / smooth;
}

// ---------------------------------------------------------------------------
extern "C" void kernel_launch(void* const* d_in, const int* in_sizes, int n_in,
                              void* d_out, int out_size, void* d_ws,
                              size_t ws_size, hipStream_t stream) {
  const float* z            = (const float*)d_in[0];
  const float* embed        = (const float*)d_in[1];
  const float* cluster_size = (const float*)d_in[2];
  const float* embed_avg    = (const float*)d_in[3];

  // outputs, flattened in return order
  float* out      = (float*)d_out;
  float* q_st     = out;                                  // N*D
  float* idx_out  = q_st + (size_t)N_ROWS * D_DIM;        // N
  float* loss_out = idx_out + N_ROWS;                     // 1
  float* new_emb  = loss_out + 1;                         // K*D
  float* ncs_out  = new_emb + (size_t)K_CODES * D_DIM;    // K
  float* navg_out = ncs_out + K_CODES;                    // K*D

  // workspace layout
  float* ws        = (float*)d_ws;
  float* enorm     = ws;                                   // K
  float* counts    = enorm + K_CODES;                      // K      (zeroed)
  float* embed_sum = counts + K_CODES;                     // K*D    (zeroed)
  float* sqsum     = embed_sum + (size_t)K_CODES * D_DIM;  // 1      (zeroed)
  float* nsum      = sqsum + 1;                            // 1
  int*   idx_ws    = (int*)(nsum + 1);                     // N

  // zero the scatter accumulators (graph-capture-safe async memset)
  hipMemsetAsync(counts, 0,
                 sizeof(float) * ((size_t)K_CODES + (size_t)K_CODES * D_DIM + 1),
                 stream);

  vq_enorm_kernel<<<K_CODES / 256, 256, 0, stream>>>(embed, enorm);
  vq_argmin_kernel<<<N_ROWS / 128, 256, 0, stream>>>(z, embed, enorm, idx_ws,
                                                     idx_out);
  vq_gather_scatter_kernel<<<N_ROWS, 256, 0, stream>>>(z, embed, idx_ws, q_st,
                                                       counts, embed_sum, sqsum);
  vq_finalize_cs_kernel<<<1, 1024, 0, stream>>>(cluster_size, counts, sqsum,
                                                ncs_out, loss_out, nsum);
  vq_finalize_embed_kernel<<<K_CODES, 256, 0, stream>>>(
      embed_avg, embed_sum, ncs_out, nsum, new_emb, navg_out);
}